// TurbulentKineticEnergy_58909771432605
// MI455X (gfx1250) — compile-verified
//
#include <hip/hip_runtime.h>
#include <hip/hip_bf16.h>
#include <math.h>

#define NX 352
#define NY 256
#define NZ 64

static const long long N3 = (long long)NX * NY * NZ;          // 5,767,168
static const long long N4 = (long long)NX * NY * NZ * 3;      // 17,301,504

// ---------------------------------------------------------------------------
// 1) Bulk copy via CDNA5 async global<->LDS path (16B per lane).
//    global_load_async_to_lds_b128 writes LDS without a VGPR round trip;
//    s_wait_asynccnt 0 orders the LDS write before the async store reads it.
// ---------------------------------------------------------------------------
__global__ __launch_bounds__(256) void copy_async_kernel(const float* __restrict__ src,
                                                         float* __restrict__ dst,
                                                         long long n16) {
    __shared__ float smem[256 * 4];
    long long gid = (long long)blockIdx.x * 256 + threadIdx.x;
    if (gid >= n16) return;
    const float* sp = src + gid * 4;
    float*       dp = dst + gid * 4;
    unsigned lds_off = (unsigned)(size_t)(&smem[threadIdx.x * 4]);
    asm volatile("global_load_async_to_lds_b128 %0, %1, off"
                 :: "v"(lds_off), "v"(sp) : "memory");
    asm volatile("s_wait_asynccnt 0x0" ::: "memory");
    asm volatile("global_store_async_from_lds_b128 %0, %1, off"
                 :: "v"(dp), "v"(lds_off) : "memory");
    // s_endpgm performs an implicit wait-idle; nothing else reads smem.
}

__global__ __launch_bounds__(256) void zero_kernel(float* __restrict__ p, long long n) {
    long long gid = (long long)blockIdx.x * 256 + threadIdx.x;
    if (gid < n) p[gid] = 0.0f;
}

// ---------------------------------------------------------------------------
// 2) Implicit vertical diffusion/dissipation: Thomas solve per column.
//    cp/dp staged in LDS, layout [z*64 + tid] (conflict-free on 64 banks).
// ---------------------------------------------------------------------------
__global__ __launch_bounds__(64) void tridiag_kernel(
    const float* __restrict__ tke, const float* __restrict__ kappaM,
    const float* __restrict__ mxl, const float* __restrict__ forc,
    const float* __restrict__ forc_surf, const int* __restrict__ kbot,
    const float* __restrict__ dzt, const float* __restrict__ dzw,
    float* out) {
    __shared__ float s_cp[NZ * 64];
    __shared__ float s_dp[NZ * 64];
    const int tid = threadIdx.x;
    const int ncols = (NX - 4) * (NY - 4);
    int col = blockIdx.x * 64 + tid;
    if (col >= ncols) return;
    const int x = 2 + col / (NY - 4);
    const int y = 2 + col % (NY - 4);
    const size_t base3 = ((size_t)x * NY + y) * NZ;
    const size_t base4 = base3 * 3;

    const int ks = kbot[x * NY + y] - 1;
    const bool land = (ks >= 0);
    const float c_eps = 0.7f;
    const float dzw_last = dzw[NZ - 1];

    float cp_prev = 0.0f, dp_prev = 0.0f;
    float kap_z = kappaM[base3];
    float delta_m = 0.0f;  // delta[z-1]

    for (int z = 0; z < NZ; ++z) {
        float kap_zp1 = (z < NZ - 1) ? kappaM[base3 + z + 1] : 0.0f;
        float delta_z = (z < NZ - 1) ? (0.5f * (kap_z + kap_zp1) / dzt[z + 1]) : 0.0f;
        float tke_tau = tke[base4 + (size_t)z * 3 + 0];
        float sq = sqrtf(fmaxf(0.0f, tke_tau));
        float mxl_z = mxl[base3 + z];
        float dzw_z = dzw[z];
        bool water = land && (z >= ks);
        bool edge = land && (z == ks);
        float a_raw, b_raw;
        if (z == 0) {
            a_raw = 0.0f; b_raw = 0.0f;
        } else if (z < NZ - 1) {
            a_raw = -delta_m / dzw_z;
            b_raw = 1.0f + (delta_z + delta_m) / dzw_z + c_eps * sq / mxl_z;
        } else {
            a_raw = -delta_m / (0.5f * dzw_last);
            b_raw = 1.0f + delta_m / (0.5f * dzw_last) + c_eps * sq / mxl_z;
        }
        float b_edge = 1.0f + delta_z / dzw_z + c_eps * sq / mxl_z;
        float c_raw = (z < NZ - 1) ? (-delta_z / dzw_z) : 0.0f;
        float d_raw = tke_tau + forc[base3 + z];
        if (z == NZ - 1) d_raw += forc_surf[x * NY + y] / (0.5f * dzw_last);

        float a = (water && !edge) ? a_raw : 0.0f;
        float b = edge ? b_edge : (water ? b_raw : 1.0f);
        float c = water ? c_raw : 0.0f;
        float d = water ? d_raw : 0.0f;

        float denom = b - a * cp_prev;
        float cp = c / denom;
        float dpv = (d - a * dp_prev) / denom;
        s_cp[z * 64 + tid] = cp;
        s_dp[z * 64 + tid] = dpv;
        cp_prev = cp; dp_prev = dpv;
        delta_m = delta_z; kap_z = kap_zp1;
    }

    // backward substitution + stage-1 write (taup1) + surf-corr
    float xnext = 0.0f;
    for (int z = NZ - 1; z >= 0; --z) {
        float sol = s_dp[z * 64 + tid] - s_cp[z * 64 + tid] * xnext;
        xnext = sol;
        bool water = land && (z >= ks);
        float st = water ? sol : tke[base4 + (size_t)z * 3 + 1];
        if (z == NZ - 1) {
            float corr = (st < 0.0f) ? (-st * 0.5f * dzw_last) : 0.0f;  // dt_tke = 1
            out[(size_t)(2 * N4) + (size_t)x * NY + y] = corr;
            st = fmaxf(0.0f, st);
        }
        out[base4 + (size_t)z * 3 + 1] = st;
    }
}

// ---------------------------------------------------------------------------
// 3) Superbee advection + horizontal diffusion + AB time step (per cell).
// ---------------------------------------------------------------------------
__device__ __forceinline__ float limiter_fn(float cr) {
    return fmaxf(0.0f, fmaxf(fminf(1.0f, 2.0f * cr), fminf(2.0f, cr)));
}

__device__ __forceinline__ float sbflux(float vel, float velfac, float dx_inv,
                                        float vm1, float v0, float vp1, float vp2,
                                        float mm1, float m0, float mp1) {
    float vv = velfac * vel;
    float uCFL = fabsf(vv * dx_inv);               // dt_tracer = 1
    float rjp = (vp2 - vp1) * mp1;
    float rj  = (vp1 - v0) * m0;
    float rjm = (v0 - vm1) * mm1;
    float div = (fabsf(rj) < 1e-20f) ? 1e-20f : rj;
    float cr = limiter_fn(((vel > 0.0f) ? rjm : rjp) / div);
    return vv * (vp1 + v0) * 0.5f - fabsf(vv) * ((1.0f - cr) + uCFL * cr) * rj * 0.5f;
}

#define T3(i, j, k) tke[(((size_t)(i) * NY + (j)) * NZ + (k)) * 3]        // tau slice
#define MW3(i, j, k) maskW[((size_t)(i) * NY + (j)) * NZ + (k)]

__global__ __launch_bounds__(256) void cell_update_kernel(
    const float* __restrict__ u, const float* __restrict__ v, const float* __restrict__ w,
    const float* __restrict__ maskU, const float* __restrict__ maskV,
    const float* __restrict__ maskW,
    const float* __restrict__ dxt, const float* __restrict__ dxu,
    const float* __restrict__ dyt, const float* __restrict__ dyu,
    const float* __restrict__ dzw, const float* __restrict__ cost,
    const float* __restrict__ cosu,
    const float* __restrict__ tke, const float* __restrict__ dtke,
    float* out) {
    long long gid = (long long)blockIdx.x * 256 + threadIdx.x;
    if (gid >= N3) return;
    const int z = (int)(gid % NZ);
    const int y = (int)((gid / NZ) % NY);
    const int x = (int)(gid / ((long long)NZ * NY));
    const size_t idx3 = (size_t)gid;
    const size_t idx4 = idx3 * 3;
    const bool interior = (x >= 2 && x < NX - 2 && y >= 2 && y < NY - 2);

    float dtk;
    float mW = maskW[idx3];

    if (interior) {
        const float cinv = 1.0f / cost[y];
        // --- adv_fe at x and x-1 (defined for i in [1, NX-3]) ---
        float fe, fem;
        {
            int i = x;
            float vel = u[(((size_t)i * NY + y) * NZ + z) * 3];
            float dinv = cinv / dxt[i];
            fe = sbflux(vel, 1.0f, dinv,
                        T3(i - 1, y, z), T3(i, y, z), T3(i + 1, y, z), T3(i + 2, y, z),
                        MW3(i, y, z) * MW3(i - 1, y, z),
                        MW3(i + 1, y, z) * MW3(i, y, z),
                        MW3(i + 2, y, z) * MW3(i + 1, y, z));
            i = x - 1;
            vel = u[(((size_t)i * NY + y) * NZ + z) * 3];
            dinv = cinv / dxt[i];
            fem = sbflux(vel, 1.0f, dinv,
                         T3(i - 1, y, z), T3(i, y, z), T3(i + 1, y, z), T3(i + 2, y, z),
                         MW3(i, y, z) * MW3(i - 1, y, z),
                         MW3(i + 1, y, z) * MW3(i, y, z),
                         MW3(i + 2, y, z) * MW3(i + 1, y, z));
        }
        // --- adv_fn at y and y-1 (defined for j in [1, NY-3]) ---
        float fn, fnm;
        {
            int j = y;
            float vel = v[(((size_t)x * NY + j) * NZ + z) * 3];
            float dinv = 1.0f / (cost[j] * dyt[j]);
            fn = sbflux(vel, cosu[j], dinv,
                        T3(x, j - 1, z), T3(x, j, z), T3(x, j + 1, z), T3(x, j + 2, z),
                        MW3(x, j, z) * MW3(x, j - 1, z),
                        MW3(x, j + 1, z) * MW3(x, j, z),
                        MW3(x, j + 2, z) * MW3(x, j + 1, z));
            j = y - 1;
            vel = v[(((size_t)x * NY + j) * NZ + z) * 3];
            dinv = 1.0f / (cost[j] * dyt[j]);
            fnm = sbflux(vel, cosu[j], dinv,
                         T3(x, j - 1, z), T3(x, j, z), T3(x, j + 1, z), T3(x, j + 2, z),
                         MW3(x, j, z) * MW3(x, j - 1, z),
                         MW3(x, j + 1, z) * MW3(x, j, z),
                         MW3(x, j + 2, z) * MW3(x, j + 1, z));
        }
        // --- adv_ft at z and z-1 (defined for k in [0, NZ-2]; k=NZ-1 is 0) ---
        float ft = 0.0f, ftm = 0.0f;
        {
            if (z < NZ - 1) {
                int k = z;
                float vel = w[(((size_t)x * NY + y) * NZ + k) * 3];
                float vm1 = (k > 0) ? T3(x, y, k - 1) : 0.0f;
                float vp2 = (k + 2 < NZ) ? T3(x, y, k + 2) : 0.0f;
                float mm1 = (k > 0) ? (MW3(x, y, k) * MW3(x, y, k - 1)) : 0.0f;
                float m0 = MW3(x, y, k + 1) * MW3(x, y, k);
                float mp1 = (k + 1 < NZ - 1) ? (MW3(x, y, k + 2) * MW3(x, y, k + 1)) : 0.0f;
                ft = sbflux(vel, 1.0f, 1.0f / dzw[k],
                            vm1, T3(x, y, k), T3(x, y, k + 1), vp2, mm1, m0, mp1);
            }
            if (z > 0) {
                int k = z - 1;  // k <= NZ-2 always
                float vel = w[(((size_t)x * NY + y) * NZ + k) * 3];
                float vm1 = (k > 0) ? T3(x, y, k - 1) : 0.0f;
                float vp2 = (k + 2 < NZ) ? T3(x, y, k + 2) : 0.0f;
                float mm1 = (k > 0) ? (MW3(x, y, k) * MW3(x, y, k - 1)) : 0.0f;
                float m0 = MW3(x, y, k + 1) * MW3(x, y, k);
                float mp1 = (k + 1 < NZ - 1) ? (MW3(x, y, k + 2) * MW3(x, y, k + 1)) : 0.0f;
                ftm = sbflux(vel, 1.0f, 1.0f / dzw[k],
                             vm1, T3(x, y, k), T3(x, y, k + 1), vp2, mm1, m0, mp1);
            }
        }
        dtk = mW * (-(fe - fem) * cinv / dxt[x] - (fn - fnm) * cinv / dyt[y]);
        if (z == 0)             dtk += -ft / dzw[0];
        else if (z < NZ - 1)    dtk += -(ft - ftm) / dzw[z];
        else                    dtk += -(0.0f - ftm) / (0.5f * dzw[NZ - 1]);
    } else {
        dtk = dtke[idx4 + 0];   // unchanged outside the interior
    }

    // tke taup1: stage-1 (tridiag result, already in out) + diffusion + AB step
    float tkep1 = out[idx4 + 1];
    if (interior) {
        const float cinv = 1.0f / cost[y];
        const float K = 2000.0f;
        float fe_d = K * (T3(x + 1, y, z) - T3(x, y, z)) * cinv / dxu[x]
                     * maskU[idx3];
        float fed_m = K * (T3(x, y, z) - T3(x - 1, y, z)) * cinv / dxu[x - 1]
                      * maskU[(((size_t)(x - 1)) * NY + y) * NZ + z];
        float fn_d = K * (T3(x, y + 1, z) - T3(x, y, z)) / dyu[y]
                     * maskV[idx3] * cosu[y];
        float fnd_m = K * (T3(x, y, z) - T3(x, y - 1, z)) / dyu[y - 1]
                      * maskV[((size_t)x * NY + (y - 1)) * NZ + z] * cosu[y - 1];
        tkep1 += mW * ((fe_d - fed_m) * cinv / dxt[x] + (fn_d - fnd_m) * cinv / dyt[y]);
    }
    tkep1 += 1.6f * dtk - 0.6f * dtke[idx4 + 2];   // (1.5+eps), (0.5+eps), eps=0.1

    out[idx4 + 1] = tkep1;            // tke taup1
    out[(size_t)N4 + idx4 + 0] = dtk; // dtke tau
}

// ---------------------------------------------------------------------------
extern "C" void kernel_launch(void* const* d_in, const int* in_sizes, int n_in,
                              void* d_out, int out_size, void* d_ws, size_t ws_size,
                              hipStream_t stream) {
    const float* u      = (const float*)d_in[0];
    const float* v      = (const float*)d_in[1];
    const float* w      = (const float*)d_in[2];
    const float* maskU  = (const float*)d_in[3];
    const float* maskV  = (const float*)d_in[4];
    const float* maskW  = (const float*)d_in[5];
    const float* dxt    = (const float*)d_in[6];
    const float* dxu    = (const float*)d_in[7];
    const float* dyt    = (const float*)d_in[8];
    const float* dyu    = (const float*)d_in[9];
    const float* dzt    = (const float*)d_in[10];
    const float* dzw    = (const float*)d_in[11];
    const float* cost   = (const float*)d_in[12];
    const float* cosu   = (const float*)d_in[13];
    const int*   kbot   = (const int*)d_in[14];
    const float* kappaM = (const float*)d_in[15];
    const float* mxl    = (const float*)d_in[16];
    const float* forc   = (const float*)d_in[17];
    const float* fts    = (const float*)d_in[18];
    const float* tke    = (const float*)d_in[19];
    const float* dtke   = (const float*)d_in[20];
    float* out = (float*)d_out;

    // 1) async bulk copy of tke and dtke into d_out (defaults / unchanged slices)
    long long n16 = N4 / 4;                       // 16B chunks per array
    int cblocks = (int)((n16 + 255) / 256);
    copy_async_kernel<<<cblocks, 256, 0, stream>>>(tke, out, n16);
    copy_async_kernel<<<cblocks, 256, 0, stream>>>(dtke, out + N4, n16);

    // 2) zero surf-corr region
    long long nsurf = (long long)NX * NY;
    zero_kernel<<<(int)((nsurf + 255) / 256), 256, 0, stream>>>(out + 2 * N4, nsurf);

    // 3) tridiagonal solve per interior column
    int ncols = (NX - 4) * (NY - 4);
    tridiag_kernel<<<(ncols + 63) / 64, 64, 0, stream>>>(
        tke, kappaM, mxl, forc, fts, kbot, dzt, dzw, out);

    // 4) fluxes + dtke + AB step per cell
    int bblocks = (int)((N3 + 255) / 256);
    cell_update_kernel<<<bblocks, 256, 0, stream>>>(
        u, v, w, maskU, maskV, maskW, dxt, dxu, dyt, dyu, dzw, cost, cosu,
        tke, dtke, out);
}